// Cluster_61744449847482
// MI455X (gfx1250) — compile-verified
//
#include <hip/hip_runtime.h>
#include <math.h>

typedef float v2f __attribute__((ext_vector_type(2)));
typedef float v8f __attribute__((ext_vector_type(8)));

#define BATCH 32
#define CIN   64
#define SDIM  56
#define HEADS 4
#define HD    24
#define INNER 96
#define TS    28          // fold tile side
#define NTOK  784         // 28*28 tokens per tile
#define CROWS 4           // chunk rows
#define CTOK  112         // 4*28 tokens per chunk (7 n-tiles of 16)
#define NCHUNK 7
#define OUTC  64

// ---- gfx1250 async global->LDS path (guarded; falls back to float4 copies) ----
#if defined(__has_builtin)
#if __has_builtin(__builtin_amdgcn_global_load_async_to_lds_b128)
#define HAVE_ASYNC_LDS 1
#endif
#if __has_builtin(__builtin_amdgcn_s_wait_asynccnt)
#define HAVE_WAIT_ASYNC 1
#endif
#endif

#if defined(HAVE_ASYNC_LDS)
typedef int v4i_t __attribute__((vector_size(16)));
typedef __attribute__((address_space(1))) v4i_t* g_v4i_p;   // global (AS1)
typedef __attribute__((address_space(3))) v4i_t* l_v4i_p;   // LDS (AS3)
__device__ static inline g_v4i_p to_gbl(const void* p) {
    return (g_v4i_p)(unsigned long long)p;
}
__device__ static inline l_v4i_p to_lds(void* p) {
    // flat LDS-aperture address: low 32 bits are the LDS byte offset (ISA 10.2)
    return (l_v4i_p)(unsigned int)(unsigned long long)p;
}
#endif

// dynamic LDS layout (floats):
//  xs 7168 | fS 10752 | simv 3136 | xq 256 | cn 384 | vc 384 | agg 384
//  pmat 1024 | cnt 16 | idxs 3136 bytes
#define SMEM_FLOATS (7168 + 10752 + 3136 + 256 + 384 + 384 + 384 + 1024 + 16)
#define SMEM_BYTES  (SMEM_FLOATS * 4 + HEADS * NTOK)

__global__ __launch_bounds__(256)
void fused_cluster_kernel(const float* __restrict__ x,
                          const float* __restrict__ Wf, const float* __restrict__ bf,
                          const float* __restrict__ Wv, const float* __restrict__ bv,
                          const float* __restrict__ Wp, const float* __restrict__ bp,
                          const float* __restrict__ alpha_p,
                          const float* __restrict__ beta_p,
                          float* __restrict__ out)
{
    extern __shared__ char smem_raw[];
    float* xs   = (float*)smem_raw;          // [64][112]
    float* fS   = xs   + CIN * CTOK;         // [96][112]
    float* simv = fS   + INNER * CTOK;       // [4][784]
    float* xq   = simv + HEADS * NTOK;       // [64][4]
    float* cn   = xq   + CIN * 4;            // [4][4][24] normalized feat centers
    float* vc   = cn   + HEADS * 4 * HD;     // [4][4][24] value centers
    float* agg  = vc   + HEADS * 4 * HD;     // [4][4][24]
    float* pmat = agg  + HEADS * 4 * HD;     // [4][64][4]
    float* cnt  = pmat + HEADS * OUTC * 4;   // [4][4]
    unsigned char* idxs = (unsigned char*)(cnt + HEADS * 4); // [4][784]

    const int t    = threadIdx.x;
    const int lane = t & 31;
    const int wave = t >> 5;
    const int blk  = blockIdx.x;
    const int batch = blk >> 2;
    const int f1 = (blk >> 1) & 1;
    const int f2 = blk & 1;
    const int row0 = f1 * TS;
    const int col0 = f2 * TS;
    const float alpha = alpha_p[0];
    const float beta  = beta_p[0];

    // ---- Phase 1: quadrant means of x (pool-before-conv fusion) ----
    {
        int c = t >> 2, m = t & 3;
        int pr = (m >> 1) * 14, pc = (m & 1) * 14;
        const float* xb = x + ((size_t)(batch * CIN + c) * SDIM + row0 + pr) * SDIM + col0 + pc;
        float acc = 0.f;
        for (int ii = 0; ii < 14; ++ii)
            #pragma unroll
            for (int jj = 0; jj < 14; ++jj)
                acc += xb[ii * SDIM + jj];
        xq[t] = acc * (1.0f / 196.0f);
    }
    __syncthreads();

    // ---- Phase 2: centers = W * pooled_x + b (tiny 96x64 @ 64x4) ----
    for (int p = t; p < INNER * 4; p += 256) {
        int oc = p >> 2, m = p & 3;
        float accf = bf[oc], accv = bv[oc];
        #pragma unroll 8
        for (int c = 0; c < CIN; ++c) {
            float xv = xq[c * 4 + m];
            accf = fmaf(Wf[oc * CIN + c], xv, accf);
            accv = fmaf(Wv[oc * CIN + c], xv, accv);
        }
        int e = oc / HD, cc = oc % HD;
        int di = (e * 4 + m) * HD + cc;
        cn[di] = accf;
        vc[di] = accv;
    }
    if (t < HEADS * 4 * HD) agg[t] = 0.f;
    if (t < HEADS * 4)      cnt[t] = 0.f;
    __syncthreads();
    if (t < HEADS * 4) {   // l2-normalize feat centers per (head, m)
        float s = 0.f;
        #pragma unroll
        for (int cc = 0; cc < HD; ++cc) { float v = cn[t * HD + cc]; s = fmaf(v, v, s); }
        float r = 1.0f / fmaxf(sqrtf(s), 1e-12f);
        #pragma unroll
        for (int cc = 0; cc < HD; ++cc) cn[t * HD + cc] *= r;
    }
    __syncthreads();

    // ---- Phase 3: stream token chunks: WMMA GEMM + clustering ----
    for (int ch = 0; ch < NCHUNK; ++ch) {
        const int rbase = row0 + ch * CROWS;
        // stage x chunk [64 ch][112 tok] into LDS, 16B granules:
        // async DMA (GLOBAL_LOAD_ASYNC_TO_LDS_B128) when available
        for (int e4 = t; e4 < (CIN * CTOK) / 4; e4 += 256) {
            int e = e4 * 4;
            int c = e / CTOK;
            int r = (e % CTOK) / TS;
            int j = e % TS;
            const float* gp = &x[((size_t)(batch * CIN + c) * SDIM + rbase + r) * SDIM + col0 + j];
            float* lp = &xs[e];
#if defined(HAVE_ASYNC_LDS)
            __builtin_amdgcn_global_load_async_to_lds_b128(to_gbl(gp), to_lds(lp), 0, 0);
#else
            *(float4*)lp = *(const float4*)gp;
#endif
        }
        if (ch + 1 < NCHUNK) {   // gfx1250 global_prefetch path for next chunk
            const float* nx = x + ((size_t)(batch * CIN) * SDIM + rbase + CROWS) * SDIM + col0 + lane;
            __builtin_prefetch(nx, 0, 1);
        }
#if defined(HAVE_WAIT_ASYNC)
        __builtin_amdgcn_s_wait_asynccnt(0);
#endif
        __syncthreads();

        // two passes through the same fS buffer: which=0 -> feat, which=1 -> value
        for (int which = 0; which < 2; ++which) {
            const float* W    = which ? Wv : Wf;
            const float* bias = which ? bv : bf;
            // GEMM: [96 x 64] @ [64 x 112] with V_WMMA_F32_16X16X4_F32
            for (int job = wave; job < 42; job += 8) {
                int mt = job / 7;
                int n0 = (job % 7) * 16;
                int arow = mt * 16 + (lane & 15);
                int klA  = (lane >> 4) << 1;       // lanes 16-31 hold K+2/K+3
                int n    = n0 + (lane & 15);
                v8f c = {};
                #pragma unroll
                for (int k = 0; k < CIN; k += 4) {
                    v2f a, b;
                    a.x = W[arow * CIN + k + klA];
                    a.y = W[arow * CIN + k + klA + 1];
                    b.x = xs[(k + klA) * CTOK + n];
                    b.y = xs[(k + klA + 1) * CTOK + n];
                    c = __builtin_amdgcn_wmma_f32_16x16x4_f32(
                            false, a, false, b, (short)0, c, false, false);
                }
                int rM = mt * 16 + ((lane >> 4) << 3);
                #pragma unroll
                for (int v = 0; v < 8; ++v) {
                    int grow = rM + v;
                    fS[grow * CTOK + n] = c[v] + bias[grow];
                }
            }
            __syncthreads();

            if (which == 0) {
                // sim + hard top-1 assignment per (token, head)
                for (int jid = t; jid < HEADS * CTOK; jid += 256) {
                    int e = jid / CTOK, tok = jid % CTOK;
                    float f[HD]; float ss = 0.f;
                    #pragma unroll
                    for (int cc = 0; cc < HD; ++cc) {
                        float v = fS[(e * HD + cc) * CTOK + tok];
                        f[cc] = v; ss = fmaf(v, v, ss);
                    }
                    float rn = 1.0f / fmaxf(sqrtf(ss), 1e-12f);
                    float best = -1.f; int bm = 0;
                    #pragma unroll
                    for (int m = 0; m < 4; ++m) {
                        float dot = 0.f;
                        #pragma unroll
                        for (int cc = 0; cc < HD; ++cc)
                            dot = fmaf(cn[(e * 4 + m) * HD + cc], f[cc], dot);
                        float z = beta + alpha * dot * rn;
                        float sg = 1.0f / (1.0f + __expf(-z));
                        if (sg > best) { best = sg; bm = m; }
                    }
                    int gn = ch * CTOK + tok;
                    simv[e * NTOK + gn] = best;
                    idxs[e * NTOK + gn] = (unsigned char)bm;
                    atomicAdd(&cnt[e * 4 + bm], 1.0f);
                }
            } else {
                // agg[m] += s_n * value[:, n] for the assigned center
                for (int jid = t; jid < HEADS * CTOK; jid += 256) {
                    int e = jid / CTOK, tok = jid % CTOK;
                    int gn = ch * CTOK + tok;
                    float s = simv[e * NTOK + gn];
                    int bm  = idxs[e * NTOK + gn];
                    float* ag = &agg[(e * 4 + bm) * HD];
                    #pragma unroll
                    for (int cc = 0; cc < HD; ++cc)
                        atomicAdd(&ag[cc], s * fS[(e * HD + cc) * CTOK + tok]);
                }
            }
            __syncthreads();
        }
    }

    // ---- Phase 4: finalize agg, fold Wp into per-center matrix P ----
    if (t < HEADS * 4 * HD) {
        int em = t / HD;
        agg[t] = (agg[t] + vc[t]) / (cnt[em] + 1.0f);
    }
    __syncthreads();
    for (int p = t; p < HEADS * OUTC * 4; p += 256) {
        int e = p >> 8, o = (p >> 2) & 63, m = p & 3;
        float s = 0.f;
        #pragma unroll
        for (int cc = 0; cc < HD; ++cc)
            s = fmaf(Wp[o * INNER + e * HD + cc], agg[(e * 4 + m) * HD + cc], s);
        pmat[p] = s;   // pmat[e][o][m]
    }
    __syncthreads();

    // ---- Phase 5: dispatch + projection collapsed to a 4-FMA gather ----
    for (int o = 0; o < OUTC; ++o) {
        float bias = bp[o];
        for (int n = t; n < NTOK; n += 256) {
            float acc = bias;
            #pragma unroll
            for (int e = 0; e < HEADS; ++e) {
                int m = idxs[e * NTOK + n];
                acc = fmaf(simv[e * NTOK + n], pmat[(e * OUTC + o) * 4 + m], acc);
            }
            int i = n / TS, j = n % TS;
            out[((size_t)(batch * OUTC + o) * SDIM + row0 + i) * SDIM + col0 + j] = acc;
        }
    }
}

extern "C" void kernel_launch(void* const* d_in, const int* in_sizes, int n_in,
                              void* d_out, int out_size, void* d_ws, size_t ws_size,
                              hipStream_t stream) {
    (void)in_sizes; (void)n_in; (void)d_ws; (void)ws_size; (void)out_size;
    const float* x  = (const float*)d_in[0];
    const float* Wf = (const float*)d_in[1];
    const float* bf = (const float*)d_in[2];
    const float* Wv = (const float*)d_in[3];
    const float* bv = (const float*)d_in[4];
    const float* Wp = (const float*)d_in[5];
    const float* bp = (const float*)d_in[6];
    const float* sa = (const float*)d_in[7];
    const float* sb = (const float*)d_in[8];
    float* out = (float*)d_out;

    (void)hipFuncSetAttribute((const void*)fused_cluster_kernel,
                              hipFuncAttributeMaxDynamicSharedMemorySize,
                              (int)SMEM_BYTES);
    fused_cluster_kernel<<<BATCH * 4, 256, SMEM_BYTES, stream>>>(
        x, Wf, bf, Wv, bv, Wp, bp, sa, sb, out);
}